// DMPNNPPoolingEdgesDirected_18906446037511
// MI455X (gfx1250) — compile-verified
//
#include <hip/hip_runtime.h>
#include <stdint.h>

#define FDIM 64
typedef float v4f __attribute__((ext_vector_type(4)));

// 64-bit payload type expected by the async-to-LDS builtin (from hipcc diagnostic:
// parameter type is 'int __attribute__((vector_size(2*sizeof(int)))) __device__ *').
typedef int v2i __attribute__((vector_size(8)));
typedef v2i __attribute__((address_space(1)))* g_v2i_p;   // global (AS1)
typedef v2i __attribute__((address_space(3)))* l_v2i_p;   // LDS   (AS3)

// ---------------- gfx1250 async-to-LDS helpers (guarded; fallbacks keep compile clean) ---
__device__ __forceinline__ void async_copy_i64_to_lds(const long long* gp, long long* lp) {
#if __has_builtin(__builtin_amdgcn_global_load_async_to_lds_b64)
    // generic->AS1 and generic->AS3 via C-style pointer casts (addrspacecast).
    __builtin_amdgcn_global_load_async_to_lds_b64(
        (g_v2i_p)gp, (l_v2i_p)lp, /*imm offset*/0, /*cpol*/0);
#else
    *lp = *gp;   // fallback: vmem load + ds_store
#endif
}

__device__ __forceinline__ void wait_async0() {
#if __has_builtin(__builtin_amdgcn_s_wait_asynccnt)
    __builtin_amdgcn_s_wait_asynccnt(0);
#else
    asm volatile("s_wait_asynccnt 0" ::: "memory");
#endif
}

// ---------------- kernel 1: zero the pooled accumulator (L2-resident, RT stores) ---------
__global__ void dmpnn_zero_kernel(float* __restrict__ pooled, size_t n4) {
    size_t i = (size_t)blockIdx.x * blockDim.x + threadIdx.x;
    if (i < n4) {
        v4f z = {0.f, 0.f, 0.f, 0.f};
        ((v4f*)pooled)[i] = z;
    }
}

// ---------------- kernel 2: scatter-add edges into pooled[dst] ---------------------------
// 256 threads = 16 groups x 16 lanes; each group owns one edge row (64 f32 = 16 x float4).
// Block tile = 64 edges; dst indices async-staged into LDS.
__global__ void dmpnn_scatter_kernel(const float* __restrict__ edges,
                                     const long long* __restrict__ dst_idx,   // edge_index row 0
                                     float* __restrict__ pooled,
                                     int n_edges) {
    __shared__ long long s_dst[64];
    const int t    = threadIdx.x;
    const int tile = blockIdx.x * 64;

    if (t < 64) {
        int e = tile + t;
        if (e < n_edges) async_copy_i64_to_lds(dst_idx + e, &s_dst[t]);
    }
    wait_async0();
    __syncthreads();

    const int group = t >> 4;        // 0..15
    const int q     = t & 15;        // float4 slot within the 64-float row
#pragma unroll
    for (int p = 0; p < 4; ++p) {
        const int et = p * 16 + group;
        const int e  = tile + et;
        if (e >= n_edges) break;
        const long long d = s_dst[et];
        // Streaming read of edges: NT so the 256MB stream doesn't evict pooled from L2.
        v4f v = __builtin_nontemporal_load((v4f*)(edges + (size_t)e * FDIM + q * 4));
        float* dp = pooled + (size_t)d * FDIM + q * 4;
        atomicAdd(dp + 0, v.x);      // global_atomic_add_f32, executes at L2
        atomicAdd(dp + 1, v.y);
        atomicAdd(dp + 2, v.z);
        atomicAdd(dp + 3, v.w);
    }
}

// ---------------- kernel 3: out[e] = pooled[src[e]] - edges[pair[e]] ---------------------
__global__ void dmpnn_gather_sub_kernel(const float* __restrict__ edges,
                                        const long long* __restrict__ src_idx,  // edge_index row 1
                                        const long long* __restrict__ pair_idx, // edge_pairs row 0
                                        const float* __restrict__ pooled,
                                        float* __restrict__ out,
                                        int n_edges) {
    __shared__ long long s_src[64];
    __shared__ long long s_pair[64];
    const int t    = threadIdx.x;
    const int tile = blockIdx.x * 64;

    if (t < 64) {
        int e = tile + t;
        if (e < n_edges) async_copy_i64_to_lds(src_idx + e, &s_src[t]);
    } else if (t < 128) {
        int e = tile + (t - 64);
        if (e < n_edges) async_copy_i64_to_lds(pair_idx + e, &s_pair[t - 64]);
    }
    wait_async0();
    __syncthreads();

    const int group = t >> 4;
    const int q     = t & 15;
#pragma unroll
    for (int p = 0; p < 4; ++p) {
        const int et = p * 16 + group;
        const int e  = tile + et;
        if (e >= n_edges) break;

        // Prefetch next pass's randomly-indexed rows (global_prefetch_b8).
        if (q == 0 && p < 3 && (e + 16) < n_edges) {
            __builtin_prefetch(pooled + (size_t)s_src[et + 16] * FDIM, 0, 3);
            __builtin_prefetch(edges  + (size_t)s_pair[et + 16] * FDIM, 0, 0);
        }

        const long long s  = s_src[et];
        const long long pr = s_pair[et];
        // pooled is 12.8MB and heavily reused -> regular (RT) load, should hit L2.
        v4f a = *(const v4f*)(pooled + (size_t)s * FDIM + q * 4);
        // reverse-edge gather walks the whole 256MB edges array once -> NT.
        v4f b = __builtin_nontemporal_load((v4f*)(edges + (size_t)pr * FDIM + q * 4));
        v4f r = a - b;
        // 256MB streaming output -> NT store, keep L2 for pooled.
        __builtin_nontemporal_store(r, (v4f*)(out + (size_t)e * FDIM + q * 4));
    }
}

// ---------------- launch ----------------------------------------------------------------
extern "C" void kernel_launch(void* const* d_in, const int* in_sizes, int n_in,
                              void* d_out, int out_size, void* d_ws, size_t ws_size,
                              hipStream_t stream) {
    // setup_inputs order: nodes f32[N,64], edges f32[M,64],
    //                     edge_index i64[2,M], edge_pairs i64[1,M]
    const float*     edges      = (const float*)d_in[1];
    const long long* edge_index = (const long long*)d_in[2];
    const long long* edge_pairs = (const long long*)d_in[3];
    float*           out        = (float*)d_out;

    const int n_nodes = in_sizes[0] / FDIM;
    const int n_edges = in_sizes[1] / FDIM;

    float* pooled = (float*)d_ws;                       // 50000*64*4 = 12.8 MB scratch

    const size_t n4 = (size_t)n_nodes * FDIM / 4;
    dmpnn_zero_kernel<<<(unsigned)((n4 + 255) / 256), 256, 0, stream>>>(pooled, n4);

    const int blocks = (n_edges + 63) / 64;
    dmpnn_scatter_kernel<<<blocks, 256, 0, stream>>>(
        edges, edge_index /* row 0 */, pooled, n_edges);

    dmpnn_gather_sub_kernel<<<blocks, 256, 0, stream>>>(
        edges, edge_index + n_edges /* row 1 */, edge_pairs, pooled, out, n_edges);
}